// VINNetwork_82514911690913
// MI455X (gfx1250) — compile-verified
//
#include <hip/hip_runtime.h>

// VIN network: B=524288 elements, 4x4 grid, K=20 max-plus value iteration,
// then vp[64] @ W[64,5] + b.  VALU-bound by the iteration (~1920 FMA/MAX per
// element); final projection done on the matrix pipe via V_WMMA_F32_16X16X4_F32.

typedef __attribute__((ext_vector_type(2))) float v2f;
typedef __attribute__((ext_vector_type(8))) float v8f;

#if defined(__HIP_DEVICE_COMPILE__) && __has_builtin(__builtin_amdgcn_wmma_f32_16x16x4_f32)
#define VIN_HAVE_WMMA_F32K4 1
#else
#define VIN_HAVE_WMMA_F32K4 0
#endif

#define WSTRIDE 68  // padded LDS stride (floats) for W^T: bank-conflict-free b64 loads

__global__ __launch_bounds__(256) void vin_vp_wmma_kernel(
    const float* __restrict__ obs,   // [B,48]
    const float* __restrict__ Wn,    // [64,5]
    const float* __restrict__ bn,    // [5]
    float* __restrict__ out)         // [B,5]
{
    __shared__ float lds_w[16 * WSTRIDE];  // W^T zero-padded to 16 cols: lds_w[col*68 + k]

    const int tid = threadIdx.x;
    const long long b = (long long)blockIdx.x * 256 + tid;

    // Stage W^T (padded) once per block.
    for (int idx = tid; idx < 16 * 64; idx += 256) {
        int col = idx & 15, row = idx >> 4;
        lds_w[col * WSTRIDE + row] = (col < 5) ? Wn[row * 5 + col] : 0.0f;
    }

    // ---- Load this element's observation row (48 floats, 12x b128) ----
    float raw[48];
    const float4* orow = reinterpret_cast<const float4*>(obs + b * 48);
#pragma unroll
    for (int q = 0; q < 12; ++q) {
        float4 f = orow[q];
        raw[4 * q + 0] = f.x; raw[4 * q + 1] = f.y;
        raw[4 * q + 2] = f.z; raw[4 * q + 3] = f.w;
    }

    float p[16], s1v[16], rt[16];
#pragma unroll
    for (int c = 0; c < 16; ++c) {
        p[c]   = 1.0f - raw[3 * c + 0];
        s1v[c] = raw[3 * c + 1];
        rt[c]  = raw[3 * c + 2];   // rout; rin = rout - 0.05
    }

    // Per-(dir,cell) bases; invalid entries dead-code-eliminated.
    const int DI[4] = {-1, 1, 0, 0};
    const int DJ[4] = {0, 0, -1, 1};
    float base[4][16];
#pragma unroll
    for (int d = 0; d < 4; ++d)
#pragma unroll
        for (int c = 0; c < 16; ++c) {
            int i = c >> 2, j = c & 3, ni = i + DI[d], nj = j + DJ[d];
            base[d][c] = (ni >= 0 && ni < 4 && nj >= 0 && nj < 4)
                             ? (rt[ni * 4 + nj] - 0.05f - rt[c]) : 0.0f;
        }

    // ---- K=20 max-plus value iteration, all in registers ----
    float v[16];
#pragma unroll
    for (int c = 0; c < 16; ++c) v[c] = 0.0f;

    for (int k = 0; k < 20; ++k) {
        float nv[16];
#pragma unroll
        for (int c = 0; c < 16; ++c) {
            int i = c >> 2, j = c & 3;
            float best = v[c];
#pragma unroll
            for (int d = 0; d < 4; ++d) {
                int ni = i + DI[d], nj = j + DJ[d];
                if (ni >= 0 && ni < 4 && nj >= 0 && nj < 4)
                    best = fmaxf(best, fmaf(p[c], v[ni * 4 + nj], base[d][c]));
            }
            nv[c] = best;
        }
#pragma unroll
        for (int c = 0; c < 16; ++c) v[c] = nv[c];
    }
    // vp[4c+0]=p[c], vp[4c+1]=s1[c], vp[4c+2]=vp[4c+3]=v[c]

    __syncthreads();  // lds_w ready

    const int lane   = tid & 31;
    const int wave   = tid >> 5;
    const int laneLo = lane & 15;
    const int hi     = lane >> 4;  // 0: K pair {0,1}; 1: K pair {2,3}

    // Fold bias into accumulator init (depends on N=laneLo only).
    float bias = bn[laneLo < 5 ? laneLo : 0];
    bias = (laneLo < 5) ? bias : 0.0f;

    v8f acc[2];

#if VIN_HAVE_WMMA_F32K4
    const float* wcol = lds_w + laneLo * WSTRIDE + 2 * hi;
#pragma unroll
    for (int t = 0; t < 2; ++t) {
        // Tile rows = batch elements of lanes t*16 .. t*16+15 of this wave.
        const int srcl = t * 16 + laneLo;
        v8f c;
#pragma unroll
        for (int r = 0; r < 8; ++r) c[r] = bias;
#pragma unroll
        for (int kk = 0; kk < 16; ++kk) {
            // Build A fragment by pulling row srcl's vp elements cross-lane:
            // a.x = vp[4kk+2hi], a.y = vp[4kk+2hi+1] of row srcl.
            float px = __shfl(p[kk],   srcl);
            float sx = __shfl(s1v[kk], srcl);
            float vx = __shfl(v[kk],   srcl);
            v2f a;
            a.x = hi ? vx : px;
            a.y = hi ? vx : sx;
            v2f bm = *reinterpret_cast<const v2f*>(wcol + 4 * kk);
            c = __builtin_amdgcn_wmma_f32_16x16x4_f32(
                    false, a, false, bm, (short)0, c, false, false);
        }
        acc[t] = c;
    }

    // D layout: reg r -> row (t*16 + hi*8 + r), col laneLo.
    if (laneLo < 5) {
#pragma unroll
        for (int t = 0; t < 2; ++t) {
            long long rbase = (long long)blockIdx.x * 256 + wave * 32 + t * 16 + hi * 8;
#pragma unroll
            for (int r = 0; r < 8; ++r)
                out[(rbase + r) * 5 + laneLo] = acc[t][r];
        }
    }
#else
    // Fallback: scalar projection (should not be taken on gfx1250).
    (void)acc; (void)wave; (void)hi;
#pragma unroll
    for (int n = 0; n < 5; ++n) {
        float s = bn[n];
#pragma unroll
        for (int c = 0; c < 16; ++c) {
            s = fmaf(p[c],   lds_w[n * WSTRIDE + 4 * c + 0], s);
            s = fmaf(s1v[c], lds_w[n * WSTRIDE + 4 * c + 1], s);
            s = fmaf(v[c],   lds_w[n * WSTRIDE + 4 * c + 2] +
                             lds_w[n * WSTRIDE + 4 * c + 3], s);
        }
        out[b * 5 + n] = s;
    }
#endif
}

extern "C" void kernel_launch(void* const* d_in, const int* in_sizes, int n_in,
                              void* d_out, int out_size, void* d_ws, size_t ws_size,
                              hipStream_t stream) {
    const float* obs = (const float*)d_in[0];  // [B,48] f32
    const float* Wn  = (const float*)d_in[1];  // [64,5] f32
    const float* bnn = (const float*)d_in[2];  // [5]    f32
    float* out = (float*)d_out;                // [B,5]  f32

    const int B = in_sizes[0] / 48;            // 524288 (multiple of 256)
    const int nblk = B / 256;
    vin_vp_wmma_kernel<<<nblk, 256, 0, stream>>>(obs, Wn, bnn, out);
}